// GaussianNLLHeatmapLoss_79731772883571
// MI455X (gfx1250) — compile-verified
//
#include <hip/hip_runtime.h>
#include <cstdint>

#define NN 384
#define CC 17
#define HH 128
#define WW 96
#define HWCNT (HH * WW)                 // 12288 elements per slice
#define TPB 256
#define V4_PER_THREAD (HWCNT / 4 / TPB) // 12 float4 chunks per thread
#define LOG_2PI 1.8378770664093456f

typedef int v4i __attribute__((ext_vector_type(4)));
typedef __attribute__((address_space(1))) v4i as1_v4i;
typedef __attribute__((address_space(3))) v4i as3_v4i;

#if defined(__has_builtin)
#if __has_builtin(__builtin_amdgcn_global_load_async_to_lds_b128)
#define HAVE_ASYNC_LDS 1
#endif
#if __has_builtin(__builtin_amdgcn_s_wait_asynccnt)
#define HAVE_WAIT_ASYNC 1
#endif
#endif

__global__ __launch_bounds__(TPB) void heatmap_nll_stage1(
    const float* __restrict__ hm, const float* __restrict__ target,
    float* __restrict__ partial)
{
    __shared__ float tile[HWCNT];   // 48 KB slice
    __shared__ float red[64];       // cross-wave reduction scratch

    const int nc = blockIdx.x;      // 0 .. N*C-1
    const int t  = threadIdx.x;
    const float4* __restrict__ g4 = (const float4*)(hm + (size_t)nc * HWCNT);
    float4* l4 = (float4*)tile;

    // ---- Stage slice into LDS (async global->LDS on gfx1250) ----
#if defined(__gfx1250__) && defined(HAVE_ASYNC_LDS)
    #pragma unroll
    for (int i = 0; i < V4_PER_THREAD; ++i) {
        const int idx = t + i * TPB;
        __builtin_amdgcn_global_load_async_to_lds_b128(
            (as1_v4i*)(uintptr_t)(g4 + idx),
            (as3_v4i*)(uint32_t)(uintptr_t)(l4 + idx),
            0, 0);
    }
#if defined(HAVE_WAIT_ASYNC)
    __builtin_amdgcn_s_wait_asynccnt(0);
#else
    asm volatile("s_wait_asynccnt 0" ::: "memory");
#endif
#else
    #pragma unroll
    for (int i = 0; i < V4_PER_THREAD; ++i) {
        const int idx = t + i * TPB;
        l4[idx] = g4[idx];
    }
#endif
    __syncthreads();

    // ---- Pass 1: max over slice ----
    float m = -3.402823466e38f;
    #pragma unroll
    for (int i = 0; i < V4_PER_THREAD; ++i) {
        float4 v = ((const float4*)tile)[t + i * TPB];
        m = fmaxf(m, fmaxf(fmaxf(v.x, v.y), fmaxf(v.z, v.w)));
    }
    #pragma unroll
    for (int off = 16; off >= 1; off >>= 1)
        m = fmaxf(m, __shfl_xor(m, off, 32));
    const int wave = t >> 5;
    if ((t & 31) == 0) red[wave] = m;
    __syncthreads();
    float mx = red[0];
    #pragma unroll
    for (int wv = 1; wv < TPB / 32; ++wv) mx = fmaxf(mx, red[wv]);
    __syncthreads();   // red reused below

    // ---- Pass 2: six weighted moment sums (centered coordinates) ----
    const float cx = 0.5f * (WW - 1);   // 47.5
    const float cy = 0.5f * (HH - 1);   // 63.5
    float s = 0.f, sx = 0.f, sy = 0.f, sxx = 0.f, syy = 0.f, sxy = 0.f;
    #pragma unroll
    for (int i = 0; i < V4_PER_THREAD; ++i) {
        const int q = t + i * TPB;
        float4 v = ((const float4*)tile)[q];
        const int idx = q * 4;                   // WW=96 divisible by 4: no row crossing
        const float yc = (float)(idx / WW) - cy;
        const float x0 = (float)(idx % WW) - cx;
        const float e0 = __expf(v.x - mx);
        const float e1 = __expf(v.y - mx);
        const float e2 = __expf(v.z - mx);
        const float e3 = __expf(v.w - mx);
        const float es  = e0 + e1 + e2 + e3;
        const float ex  = e0 * x0 + e1 * (x0 + 1.f) + e2 * (x0 + 2.f) + e3 * (x0 + 3.f);
        const float exx = e0 * x0 * x0 + e1 * (x0 + 1.f) * (x0 + 1.f)
                        + e2 * (x0 + 2.f) * (x0 + 2.f) + e3 * (x0 + 3.f) * (x0 + 3.f);
        s   += es;
        sx  += ex;
        sxx += exx;
        sy  += yc * es;
        syy += yc * yc * es;
        sxy += yc * ex;
    }
    #pragma unroll
    for (int off = 16; off >= 1; off >>= 1) {
        s   += __shfl_xor(s,   off, 32);
        sx  += __shfl_xor(sx,  off, 32);
        sy  += __shfl_xor(sy,  off, 32);
        sxx += __shfl_xor(sxx, off, 32);
        syy += __shfl_xor(syy, off, 32);
        sxy += __shfl_xor(sxy, off, 32);
    }
    if ((t & 31) == 0) {
        red[wave * 8 + 0] = s;   red[wave * 8 + 1] = sx;  red[wave * 8 + 2] = sy;
        red[wave * 8 + 3] = sxx; red[wave * 8 + 4] = syy; red[wave * 8 + 5] = sxy;
    }
    __syncthreads();

    // ---- Finish: scalar NLL for this (n,c) ----
    if (t == 0) {
        float S = 0.f, SX = 0.f, SY = 0.f, SXX = 0.f, SYY = 0.f, SXY = 0.f;
        #pragma unroll
        for (int wv = 0; wv < TPB / 32; ++wv) {
            S   += red[wv * 8 + 0]; SX  += red[wv * 8 + 1]; SY  += red[wv * 8 + 2];
            SXX += red[wv * 8 + 3]; SYY += red[wv * 8 + 4]; SXY += red[wv * 8 + 5];
        }
        const float z    = S;
        const float inv  = 1.f / z;
        const float mxc  = SX * inv;            // centered means
        const float myc  = SY * inv;
        const float x_mean = cx + mxc;
        const float y_mean = cy + myc;
        const float x_var  = (1.f / 12.f) + SXX * inv - mxc * mxc;
        const float y_var  = (1.f / 12.f) + SYY * inv - myc * myc;
        const float covar  = SXY * inv - mxc * myc;

        const float exp_max = __expf(mx - (float)HH - (float)WW);
        const float ez = exp_max * z;
        const float presence = 1.f - 1.f / (ez + 1.f);

        const float* tg = target + (size_t)nc * 3;
        const float gx = tg[0], gy = tg[1], gv = tg[2];
        const bool  msk = (gv != 0.f);
        const float label_loss = -(msk ? __logf(presence) : __logf(1.f - presence));

        const float dx  = gx - x_mean;
        const float dy  = gy - y_mean;
        const float det = x_var * y_var - covar * covar;
        const float qv  = (y_var * dx * dx - 2.f * covar * dx * dy + x_var * dy * dy) / det;
        const float mf  = msk ? 1.f : 0.f;

        partial[nc] = mf * (__logf(det) + qv) * 0.5f + LOG_2PI + label_loss;
    }
}

__global__ void heatmap_nll_stage2(const float* __restrict__ partial,
                                   float* __restrict__ out)
{
    const int n = blockIdx.x * blockDim.x + threadIdx.x;
    if (n < NN) {
        float a = 0.f;
        #pragma unroll
        for (int c = 0; c < CC; ++c) a += partial[n * CC + c];
        out[n] = a;
    }
}

extern "C" void kernel_launch(void* const* d_in, const int* in_sizes, int n_in,
                              void* d_out, int out_size, void* d_ws, size_t ws_size,
                              hipStream_t stream) {
    const float* hm     = (const float*)d_in[0];   // (N,C,H,W) f32
    const float* target = (const float*)d_in[1];   // (N,C,3)  f32
    // d_in[2] target_weight is unused by the reference computation
    float* partial = (float*)d_ws;                 // N*C floats of scratch

    heatmap_nll_stage1<<<NN * CC, TPB, 0, stream>>>(hm, target, partial);
    heatmap_nll_stage2<<<(NN + 127) / 128, 128, 0, stream>>>(partial, (float*)d_out);
}